// MetapathConv_56573309223581
// MI455X (gfx1250) — compile-verified
//
#include <hip/hip_runtime.h>

typedef __attribute__((ext_vector_type(2))) float v2f;
typedef __attribute__((ext_vector_type(8))) float v8f;

#define FD 128  // feature dim

__device__ __forceinline__ void fadd_atomic(float* p, float v) {
  // relaxed agent-scope fetch_add -> hardware global_atomic_add_f32 (no CAS loop)
  __hip_atomic_fetch_add(p, v, __ATOMIC_RELAXED, __HIP_MEMORY_SCOPE_AGENT);
}

// ---------------- utility kernels ----------------

__global__ void k_zero(float* __restrict__ p, long long n) {
  long long i = (long long)blockIdx.x * blockDim.x + threadIdx.x;
  long long stride = (long long)gridDim.x * blockDim.x;
  for (; i < n; i += stride) p[i] = 0.0f;
}

// per-(rel,row) edge counts, as float (exact up to 2^24; E=5e5)
__global__ void k_count(const int* __restrict__ row, const int* __restrict__ et,
                        float* __restrict__ cnt, int E, int N) {
  int e = blockIdx.x * blockDim.x + threadIdx.x;
  if (e >= E) return;
  fadd_atomic(&cnt[(long long)et[e] * N + row[e]], 1.0f);
}

__global__ void k_invert(float* __restrict__ cnt, long long n) {
  long long i = (long long)blockIdx.x * blockDim.x + threadIdx.x;
  long long stride = (long long)gridDim.x * blockDim.x;
  for (; i < n; i += stride) cnt[i] = 1.0f / fmaxf(cnt[i], 1.0f);
}

// ---------------- scatter hop: one wave32 per edge, 4 floats per lane -------
// e is wave-uniform: readfirstlane it so et/row/col become scalar (s_load)
// fetches instead of 32-lane VMEM gathers.

__global__ void k_hop(const float* __restrict__ cur, float* __restrict__ nxt,
                      const int* __restrict__ row, const int* __restrict__ col,
                      const int* __restrict__ et, int rel, int E) {
  int gid  = blockIdx.x * blockDim.x + threadIdx.x;
  int lane = threadIdx.x & 31;
  int e    = __builtin_amdgcn_readfirstlane(gid >> 5);   // wave-uniform edge id
  if (e >= E) return;
  if (et[e] != rel) return;                              // scalar load + branch
  int c = col[e], r = row[e];                            // scalar loads
  const float4 v = *(const float4*)(cur + (long long)c * FD + lane * 4);
  float* dst = nxt + (long long)r * FD + lane * 4;
  fadd_atomic(dst + 0, v.x);
  fadd_atomic(dst + 1, v.y);
  fadd_atomic(dst + 2, v.z);
  fadd_atomic(dst + 3, v.w);
}

// row-scale by 1/count  (inv points at the rel-specific table)
__global__ void k_norm(float* __restrict__ buf, const float* __restrict__ inv,
                       long long n) {
  long long i = (long long)blockIdx.x * blockDim.x + threadIdx.x;
  long long stride = (long long)gridDim.x * blockDim.x;
  for (; i < n; i += stride) buf[i] *= inv[i >> 7];   // i / FD
}

// ---------------- fp32 WMMA GEMM: out += (h @ W^T + b) / 3 ------------------
// One wave per 16x16 output tile. K=128 via 32x v_wmma_f32_16x16x4_f32.
// A (16x4 f32): lane L holds M=L&15, K = 2*(L>>4) + {0,1}  -> v2f
// B (4x16 f32): lane L holds N=L&15, K = 2*(L>>4) + {0,1}  -> v2f
//   B[k][n] = W[n][k]  (W row-major [out_dim][in_dim], so contiguous in k)
// C/D (16x16 f32): VGPR j, lane L -> M = j + 8*(L>>4), N = L&15
__global__ void k_gemm_wmma(const float* __restrict__ h, const float* __restrict__ Wm,
                            const float* __restrict__ bias, float* __restrict__ out,
                            int N) {
  int lane = threadIdx.x & 31;
  int wave = threadIdx.x >> 5;        // 0..7 -> column tile of 16 (FD=128)
  int row0 = blockIdx.x * 16;
  int col0 = wave * 16;
  int m    = lane & 15;
  int kk   = (lane >> 4) * 2;

  int arow = row0 + m;
  if (arow >= N) arow = N - 1;        // safe (dup) read for ragged tail
  const float* ap = h  + (long long)arow        * FD + kk;
  const float* bp = Wm + (long long)(col0 + m)  * FD + kk;

  v8f c = {};
  #pragma unroll
  for (int k = 0; k < FD; k += 4) {
    v2f a = *(const v2f*)(ap + k);    // 8B-aligned: kk even, k%4==0
    v2f b = *(const v2f*)(bp + k);
    c = __builtin_amdgcn_wmma_f32_16x16x4_f32(
        /*neg_a=*/false, a, /*neg_b=*/false, b,
        /*c_mod=*/(short)0, c, /*reuse_a=*/false, /*reuse_b=*/false);
  }

  const float s = 1.0f / 3.0f;
  int ocol = col0 + m;
  float bv = bias[ocol];
  int obase = row0 + 8 * (lane >> 4);
  #pragma unroll
  for (int j = 0; j < 8; ++j) {
    int orow = obase + j;
    if (orow < N) {
      float* o = out + (long long)orow * FD + ocol;
      *o += (c[j] + bv) * s;          // exactly one writer per element
    }
  }
}

// ---------------- orchestration ----------------

extern "C" void kernel_launch(void* const* d_in, const int* in_sizes, int n_in,
                              void* d_out, int out_size, void* d_ws, size_t ws_size,
                              hipStream_t stream) {
  const float* x   = (const float*)d_in[0];   // [N,128]
  const float* W   = (const float*)d_in[1];   // [3,128,128]
  const float* b   = (const float*)d_in[2];   // [3,128]
  const int*   ei  = (const int*)d_in[3];     // [2,E]
  const int*   et  = (const int*)d_in[4];     // [E]
  int E = in_sizes[4];
  int N = in_sizes[0] / FD;
  const int* row = ei;
  const int* col = ei + E;

  float* out  = (float*)d_out;
  float* buf0 = (float*)d_ws;                       // [N,128]
  float* buf1 = buf0 + (long long)N * FD;           // [N,128]
  float* inv  = buf1 + (long long)N * FD;           // [3,N]

  const int ZB = 2048, T = 256;
  long long ND = (long long)N * FD;

  // counts -> 1/max(cnt,1)
  k_zero  <<<ZB, T, 0, stream>>>(inv, 3LL * N);
  k_count <<<(E + T - 1) / T, T, 0, stream>>>(row, et, inv, E, N);
  k_invert<<<ZB, T, 0, stream>>>(inv, 3LL * N);

  // out = 0
  k_zero<<<ZB, T, 0, stream>>>(out, ND);

  int hopBlocks = (E * 32 + T - 1) / T;   // one wave per edge
  auto hop = [&](const float* src, float* dst, int rel) {
    k_zero<<<ZB, T, 0, stream>>>(dst, ND);
    k_hop <<<hopBlocks, T, 0, stream>>>(src, dst, row, col, et, rel, E);
    k_norm<<<ZB, T, 0, stream>>>(dst, inv + (long long)rel * N, ND);
  };

  int gemmBlocks = (N + 15) / 16;
  auto gemm = [&](const float* h, int i) {
    k_gemm_wmma<<<gemmBlocks, T, 0, stream>>>(h, W + (long long)i * FD * FD,
                                              b + (long long)i * FD, out, N);
  };

  // metapath 0: [0,1]   and reuse a = P0(x) for metapath 2: [0,2,1]
  hop(x,    buf0, 0);        // a  = P0(x)
  hop(buf0, buf1, 1);        // h0 = P1(a)
  gemm(buf1, 0);
  hop(buf0, buf1, 2);        // c  = P2(a)
  hop(buf1, buf0, 1);        // h2 = P1(c)
  gemm(buf0, 2);
  // metapath 1: [1,2]
  hop(x,    buf1, 1);        // P1(x)
  hop(buf1, buf0, 2);        // h1 = P2(P1(x))
  gemm(buf0, 1);
}